// GAT_55344948576383
// MI455X (gfx1250) — compile-verified
//
#include <hip/hip_runtime.h>

typedef __attribute__((ext_vector_type(2))) float v2f;
typedef __attribute__((ext_vector_type(8))) float v8f;

#define N_NODES 25000
#define N_EDGES 400000

// ---------------------------------------------------------------------------
// WMMA fp32 GEMM:  Z[N, Dout] = X[N, K] * W[Dout, K]^T
// One wave computes a 16 x (16*NT) strip using V_WMMA_F32_16X16X4_F32.
// f32 A (16x4): lanes 0-15 hold M=lane,K=k+{0,1}; lanes 16-31 hold K=k+{2,3}.
// f32 B (4x16): lanes 0-15 hold N=lane,K=k+{0,1}; lanes 16-31 hold K=k+{2,3}.
// C/D (16x16):  VGPR r: lanes 0-15 -> (M=r,N=lane); lanes 16-31 -> (M=r+8).
// ---------------------------------------------------------------------------
template <int K, int NT>
__global__ __launch_bounds__(128) void wmma_gemm_xwt(
    const float* __restrict__ X,   // [N, K]
    const float* __restrict__ W,   // [Dout, K]
    float* __restrict__ Z,         // [N, Dout]
    int N, int Dout) {
  const int lane = threadIdx.x & 31;
  const int wave = threadIdx.x >> 5;
  const int mtile = blockIdx.x * 4 + wave;
  const int row0 = mtile * 16;
  if (row0 >= N) return;
  const int col0 = blockIdx.y * (16 * NT);

  const int half = lane >> 4;   // which 16-lane half
  const int l15 = lane & 15;

  int arow = row0 + l15;
  if (arow >= N) arow = N - 1;  // clamp loads; stores are guarded below
  const float* aptr = X + (size_t)arow * K + 2 * half;
  const float* bptr = W + (size_t)(col0 + l15) * K + 2 * half;

  v8f c[NT] = {};

  for (int k = 0; k < K; k += 4) {
    v2f a = *(const v2f*)(aptr + k);
#pragma unroll
    for (int t = 0; t < NT; ++t) {
      v2f b = *(const v2f*)(bptr + (size_t)t * 16 * K + k);
      c[t] = __builtin_amdgcn_wmma_f32_16x16x4_f32(
          /*neg_a=*/false, a, /*neg_b=*/false, b,
          /*c_mod=*/(short)0, c[t], /*reuse_a=*/false, /*reuse_b=*/false);
    }
  }

#pragma unroll
  for (int t = 0; t < NT; ++t) {
#pragma unroll
    for (int r = 0; r < 8; ++r) {
      int row = row0 + r + 8 * half;
      if (row < N)
        Z[(size_t)row * Dout + col0 + t * 16 + l15] = c[t][r];
    }
  }
}

// ---------------------------------------------------------------------------
// Per-node attention coefficients: el[n,h] = <z[n,h,:], al[h,:]>, same for er.
// ---------------------------------------------------------------------------
template <int H, int F>
__global__ void attn_coef(const float* __restrict__ Z,
                          const float* __restrict__ al,
                          const float* __restrict__ ar,
                          float* __restrict__ el, float* __restrict__ er,
                          int N) {
  int t = blockIdx.x * blockDim.x + threadIdx.x;
  if (t >= N * H) return;
  int n = t / H, h = t % H;
  const float* zrow = Z + (size_t)n * H * F + h * F;
  const float* alr = al + h * F;
  const float* arr = ar + h * F;
  float sl = 0.f, sr = 0.f;
#pragma unroll 8
  for (int f = 0; f < F; ++f) {
    float z = zrow[f];
    sl += z * alr[f];
    sr += z * arr[f];
  }
  el[t] = sl;
  er[t] = sr;
}

// ---------------------------------------------------------------------------
// CSR row offsets from sorted dst: rs[n] = lower_bound(dst, n), rs[N] = E.
// ---------------------------------------------------------------------------
__global__ void row_offsets(const int* __restrict__ dst, int* __restrict__ rs,
                            int E, int N) {
  int n = blockIdx.x * blockDim.x + threadIdx.x;
  if (n > N) return;
  int lo = 0, hi = E;
  while (lo < hi) {
    int mid = (lo + hi) >> 1;
    if (dst[mid] < n) lo = mid + 1; else hi = mid;
  }
  rs[n] = lo;
}

// ---------------------------------------------------------------------------
// Segment softmax + weighted aggregation. One wave32 per destination node.
//   pass1: per-head max over incident edges (lanes = (slot, head), xor-reduce)
//   pass2: den[h] += exp(e-m); acc[h,f] += exp(e-m) * z[src,h,f]
//   final: out = (acc/den), optional ELU. Zero atomics; one write per element.
// ---------------------------------------------------------------------------
template <int H, int F, bool ELU>
__global__ __launch_bounds__(256) void gat_aggregate(
    const float* __restrict__ Z,    // [N, H*F]
    const float* __restrict__ el,   // [N, H]
    const float* __restrict__ er,   // [N, H]
    const int* __restrict__ src,    // [E]
    const int* __restrict__ rs,     // [N+1]
    float* __restrict__ Out,        // [N, H*F]
    int N) {
  constexpr int HF = H * F;
  constexpr int VPT = HF / 32;  // output floats per lane
  const int lane = threadIdx.x & 31;
  const int node = (blockIdx.x * blockDim.x + threadIdx.x) >> 5;
  if (node >= N) return;
  const int start = rs[node];
  const int end = rs[node + 1];
  const int h = lane % H;

  if (start >= end) {  // no incoming edges -> zeros (matches reference)
#pragma unroll
    for (int j = 0; j < VPT; ++j)
      Out[(size_t)node * HF + j * 32 + lane] = 0.f;
    return;
  }

  const float er_h = er[node * H + h];

  // pass 1: per-head running max, SLOTS edges in parallel
  constexpr int SLOTS = 32 / H;
  const int slot = lane / H;
  float m = -__builtin_inff();
  for (int e = start + slot; e < end; e += SLOTS) {
    float v = el[src[e] * H + h] + er_h;
    v = (v > 0.f) ? v : 0.2f * v;  // leaky_relu(0.2)
    m = fmaxf(m, v);
  }
#pragma unroll
  for (int off = H; off < 32; off <<= 1)
    m = fmaxf(m, __shfl_xor(m, off, 32));
  // now every lane holds m for head (lane % H)

  float den = 0.f;
  if constexpr (VPT % 4 == 0) {
    float4 acc[VPT / 4];
#pragma unroll
    for (int j = 0; j < VPT / 4; ++j) acc[j] = make_float4(0.f, 0.f, 0.f, 0.f);
    for (int e = start; e < end; ++e) {
      const int s = src[e];
      float v = el[s * H + h] + er_h;
      v = (v > 0.f) ? v : 0.2f * v;
      const float ex = expf(v - m);
      den += ex;
      const float4* zr = (const float4*)(Z + (size_t)s * HF);
#pragma unroll
      for (int j = 0; j < VPT / 4; ++j) {
        const int idx4 = j * 32 + lane;          // float4 index within row
        const int hh = (idx4 * 4) / F;           // head of this float4
        const float w = __shfl(ex, hh, 32);      // lane hh holds head hh
        const float4 zv = zr[idx4];
        acc[j].x += w * zv.x;
        acc[j].y += w * zv.y;
        acc[j].z += w * zv.z;
        acc[j].w += w * zv.w;
      }
    }
    float4* orow = (float4*)(Out + (size_t)node * HF);
#pragma unroll
    for (int j = 0; j < VPT / 4; ++j) {
      const int idx4 = j * 32 + lane;
      const int hh = (idx4 * 4) / F;
      const float d = __shfl(den, hh, 32);
      float4 o;
      o.x = acc[j].x / d; o.y = acc[j].y / d;
      o.z = acc[j].z / d; o.w = acc[j].w / d;
      if constexpr (ELU) {
        o.x = (o.x > 0.f) ? o.x : expf(o.x) - 1.f;
        o.y = (o.y > 0.f) ? o.y : expf(o.y) - 1.f;
        o.z = (o.z > 0.f) ? o.z : expf(o.z) - 1.f;
        o.w = (o.w > 0.f) ? o.w : expf(o.w) - 1.f;
      }
      orow[idx4] = o;
    }
  } else {
    float acc[VPT];
#pragma unroll
    for (int j = 0; j < VPT; ++j) acc[j] = 0.f;
    for (int e = start; e < end; ++e) {
      const int s = src[e];
      float v = el[s * H + h] + er_h;
      v = (v > 0.f) ? v : 0.2f * v;
      const float ex = expf(v - m);
      den += ex;
      const float* zr = Z + (size_t)s * HF;
#pragma unroll
      for (int j = 0; j < VPT; ++j) {
        const int idx = j * 32 + lane;
        const int hh = idx / F;
        const float w = __shfl(ex, hh, 32);
        acc[j] += w * zr[idx];
      }
    }
#pragma unroll
    for (int j = 0; j < VPT; ++j) {
      const int idx = j * 32 + lane;
      const int hh = idx / F;
      const float d = __shfl(den, hh, 32);
      float o = acc[j] / d;
      if constexpr (ELU) o = (o > 0.f) ? o : expf(o) - 1.f;
      Out[(size_t)node * HF + idx] = o;
    }
  }
}

// ---------------------------------------------------------------------------
extern "C" void kernel_launch(void* const* d_in, const int* in_sizes, int n_in,
                              void* d_out, int out_size, void* d_ws,
                              size_t ws_size, hipStream_t stream) {
  const float* h   = (const float*)d_in[0];
  const int*   src = (const int*)d_in[1];
  const int*   dst = (const int*)d_in[2];
  const float* W1  = (const float*)d_in[3];
  const float* al1 = (const float*)d_in[4];
  const float* ar1 = (const float*)d_in[5];
  const float* W2  = (const float*)d_in[6];
  const float* al2 = (const float*)d_in[7];
  const float* ar2 = (const float*)d_in[8];
  const float* W3  = (const float*)d_in[9];
  const float* al3 = (const float*)d_in[10];
  const float* ar3 = (const float*)d_in[11];

  char* ws = (char*)d_ws;
  size_t off = 0;
  auto alloc = [&](size_t bytes) -> void* {
    void* p = ws + off;
    off = (off + bytes + 255) & ~(size_t)255;
    return p;
  };
  float* bufA = (float*)alloc((size_t)N_NODES * 512 * sizeof(float)); // z buffers
  float* bufB = (float*)alloc((size_t)N_NODES * 512 * sizeof(float)); // x buffers
  float* el   = (float*)alloc((size_t)N_NODES * 8 * sizeof(float));
  float* er   = (float*)alloc((size_t)N_NODES * 8 * sizeof(float));
  int*   rs   = (int*)alloc((size_t)(N_NODES + 1) * sizeof(int));

  // CSR offsets from sorted dst
  row_offsets<<<(N_NODES + 1 + 255) / 256, 256, 0, stream>>>(dst, rs, N_EDGES,
                                                             N_NODES);

  const int MT = (N_NODES + 15) / 16;  // 1563 M-tiles
  const dim3 gemmBlk(128);
  const int aggBlocks = (N_NODES * 32 + 255) / 256;

  // ---- Layer 1: h[.,256] -> z1(bufA)[.,512] -> x1(bufB)
  wmma_gemm_xwt<256, 4><<<dim3((MT + 3) / 4, 8), gemmBlk, 0, stream>>>(
      h, W1, bufA, N_NODES, 512);
  attn_coef<8, 64><<<(N_NODES * 8 + 255) / 256, 256, 0, stream>>>(
      bufA, al1, ar1, el, er, N_NODES);
  gat_aggregate<8, 64, true><<<aggBlocks, 256, 0, stream>>>(
      bufA, el, er, src, rs, bufB, N_NODES);

  // ---- Layer 2: x1(bufB) -> z2(bufA) -> x2(bufB)
  wmma_gemm_xwt<512, 4><<<dim3((MT + 3) / 4, 8), gemmBlk, 0, stream>>>(
      bufB, W2, bufA, N_NODES, 512);
  attn_coef<8, 64><<<(N_NODES * 8 + 255) / 256, 256, 0, stream>>>(
      bufA, al2, ar2, el, er, N_NODES);
  gat_aggregate<8, 64, true><<<aggBlocks, 256, 0, stream>>>(
      bufA, el, er, src, rs, bufB, N_NODES);

  // ---- Layer 3: x2(bufB) -> z3(bufA)[.,32] -> out (no ELU)
  wmma_gemm_xwt<512, 2><<<dim3((MT + 3) / 4, 1), gemmBlk, 0, stream>>>(
      bufB, W3, bufA, N_NODES, 32);
  attn_coef<1, 32><<<(N_NODES + 255) / 256, 256, 0, stream>>>(
      bufA, al3, ar3, el, er, N_NODES);
  gat_aggregate<1, 32, false><<<aggBlocks, 256, 0, stream>>>(
      bufA, el, er, src, rs, (float*)d_out, N_NODES);
}